// PatNNUEv1_83640193122486
// MI455X (gfx1250) — compile-verified
//
#include <hip/hip_runtime.h>

// ---------------------------------------------------------------------------
// PatNNUE fused kernel for MI455X (gfx1250, wave32)
//   - embedding table (152KB) DMA'd into LDS with GLOBAL_LOAD_ASYNC_TO_LDS_B128
//     (ASYNCcnt), overlapped with the per-wave input-stream fetch
//   - 1 wave per batch element, lane = board position
//   - value-head MLP (K=4) on V_WMMA_F32_16X16X4_F32, 16 batch rows per block
// ---------------------------------------------------------------------------

#define PCODE    2380
#define NBATCH   32768
#define HWDIM    15
#define NPOS     225                      // 15*15
#define EMB_ROWS (2*(PCODE+1))            // 4762
#define TAB_FLOATS (EMB_ROWS*8)           // 38096 floats = 152384 B
#define POL_PITCH 232                     // padded 225 -> avoid LDS bank conflicts
#define WAVES    16
#define NTHREADS (WAVES*32)               // 512 threads = 16 wave32
#define NBLK     (NBATCH/WAVES)           // 2048 blocks

#define SMEM_FLOATS (TAB_FLOATS + WAVES*4*POL_PITCH + WAVES*4 + WAVES*4)
#define SMEM_BYTES  (SMEM_FLOATS * 4)     // 212288 B < 320KB LDS

typedef __attribute__((ext_vector_type(2))) float v2f;
typedef __attribute__((ext_vector_type(8))) float v8f;
typedef __attribute__((address_space(3))) void lds_void;

__device__ __forceinline__ float lrelu16(float x) {
    return x >= 0.0f ? x : x * 0.0625f;
}

// Async DMA of 16B/lane from global to LDS, tracked with ASYNCcnt.
__device__ __forceinline__ void async_copy_b128(const void* g, void* l) {
    unsigned lds_addr = (unsigned)(unsigned long long)(lds_void*)l;
    unsigned long long gaddr = (unsigned long long)g;
    asm volatile("global_load_async_to_lds_b128 %0, %1, off"
                 :: "v"(lds_addr), "v"(gaddr));
}

__device__ __forceinline__ void wait_asynccnt0() {
    asm volatile("s_wait_asynccnt 0x0" ::: "memory");
}

extern "C" __global__ __launch_bounds__(NTHREADS)
void patnnue_fused(const int* __restrict__ sparse,     // (B,12,15,15) i32
                   const int* __restrict__ board,      // (B,2,15,15)  i32
                   const float* __restrict__ W_emb,    // (4762,8)
                   const float* __restrict__ dw_w,     // (4,1,3,3)
                   const float* __restrict__ dw_b,     // (4,)
                   const float* __restrict__ pf_w,     // (1,4,1,1)
                   const float* __restrict__ l1_w,     // (4,4)
                   const float* __restrict__ l1_b,     // (4,)
                   const float* __restrict__ l2_w,     // (4,4)
                   const float* __restrict__ l2_b,     // (4,)
                   const float* __restrict__ vf_w,     // (3,4)
                   const float* __restrict__ vf_b,     // (3,)
                   float* __restrict__ v_out,          // (B,3)
                   float* __restrict__ p_out)          // (B,1,15,15)
{
    extern __shared__ float smem[];
    float* tab    = smem;                              // 38096 f
    float* pol    = tab + TAB_FLOATS;                  // 16*4*232 f
    float* vstage = pol + WAVES*4*POL_PITCH;           // 16*4 f
    float* hstage = vstage + WAVES*4;                  // 16*4 f

    const int tid  = threadIdx.x;
    const int wave = tid >> 5;
    const int lane = tid & 31;
    const int b    = blockIdx.x * WAVES + wave;

    const int* sp10 = sparse + ((size_t)b * 12 + 10) * NPOS;
    const int* sp11 = sp10 + NPOS;
    const int* bd0  = board + (size_t)b * 2 * NPOS;
    const int* bd1  = bd0 + NPOS;

    // ---- 1) kick off the async DMA of the embedding table into LDS
    {
        const float4* g = (const float4*)W_emb;
        float4*       t = (float4*)tab;
        for (int i = tid; i < TAB_FLOATS / 4; i += NTHREADS)
            async_copy_b128(g + i, t + i);
    }

    // ---- 2) while the DMA is in flight: stream this wave's inputs and
    //         resolve the embedding row indices into registers
    int idx0[8], idx1[8];
    #pragma unroll
    for (int it = 0; it < 8; ++it) {
        int pos = it * 32 + lane;
        int c0 = 0, c1 = 0;
        if (pos < NPOS) {
            c0 = sp10[pos];
            c1 = sp11[pos];
            int ne = (bd0[pos] + bd1[pos]) > 0;
            if (ne) { c0 = PCODE; c1 = PCODE; }
        }
        idx0[it] = c0 * 8;
        idx1[it] = (c1 + PCODE + 1) * 8;
    }

    wait_asynccnt0();          // our slice of the table is in LDS
    __syncthreads();           // everyone's slice is in LDS

    float* polw = pol + wave * 4 * POL_PITCH;

    // ---- 3) LDS gather + channel sum; stage policy feats, accumulate value feats
    float va0 = 0.f, va1 = 0.f, va2 = 0.f, va3 = 0.f;
    #pragma unroll
    for (int it = 0; it < 8; ++it) {
        int pos = it * 32 + lane;
        if (pos < NPOS) {
            const float4* r0 = (const float4*)(tab + idx0[it]);
            const float4* r1 = (const float4*)(tab + idx1[it]);
            float4 lo0 = r0[0], hi0 = r0[1];
            float4 lo1 = r1[0], hi1 = r1[1];

            polw[0 * POL_PITCH + pos] = lo0.x + lo1.x;
            polw[1 * POL_PITCH + pos] = lo0.y + lo1.y;
            polw[2 * POL_PITCH + pos] = lo0.z + lo1.z;
            polw[3 * POL_PITCH + pos] = lo0.w + lo1.w;

            va0 += hi0.x + hi1.x;
            va1 += hi0.y + hi1.y;
            va2 += hi0.z + hi1.z;
            va3 += hi0.w + hi1.w;
        }
    }

    // wave32 butterfly reduction of the value-head spatial sums
    #pragma unroll
    for (int m = 16; m >= 1; m >>= 1) {
        va0 += __shfl_xor(va0, m);
        va1 += __shfl_xor(va1, m);
        va2 += __shfl_xor(va2, m);
        va3 += __shfl_xor(va3, m);
    }
    if (lane == 0) {
        vstage[wave * 4 + 0] = va0;
        vstage[wave * 4 + 1] = va1;
        vstage[wave * 4 + 2] = va2;
        vstage[wave * 4 + 3] = va3;
    }

    // ---- 4) policy head: depthwise 3x3 + ReLU + pointwise 1x1 (per-wave LDS tile)
    float dwc[4][9], dbb[4], pw[4];
    #pragma unroll
    for (int c = 0; c < 4; ++c) {
        dbb[c] = dw_b[c];
        pw[c]  = pf_w[c];
        #pragma unroll
        for (int k = 0; k < 9; ++k) dwc[c][k] = dw_w[c * 9 + k];
    }
    float* pouts = p_out + (size_t)b * NPOS;
    for (int pos = lane; pos < NPOS; pos += 32) {
        int y = pos / HWDIM;
        int x = pos - y * HWDIM;
        float acc = 0.f;
        #pragma unroll
        for (int c = 0; c < 4; ++c) {
            float s = dbb[c];
            #pragma unroll
            for (int dy = -1; dy <= 1; ++dy) {
                int yy = y + dy;
                #pragma unroll
                for (int dx = -1; dx <= 1; ++dx) {
                    int xx = x + dx;
                    if ((unsigned)yy < (unsigned)HWDIM && (unsigned)xx < (unsigned)HWDIM)
                        s += dwc[c][(dy + 1) * 3 + (dx + 1)] *
                             polw[c * POL_PITCH + yy * HWDIM + xx];
                }
            }
            s = fmaxf(s, 0.f);
            acc += pw[c] * s;
        }
        pouts[pos] = acc;
    }

    __syncthreads();   // vstage complete across all 16 waves

    // ---- 5) value head: 16 batch rows, K=4 per layer -> V_WMMA_F32_16X16X4_F32
    // A layout (16x4 f32): lanes 0-15 {v0=K0,v1=K1}, lanes 16-31 {v0=K2,v1=K3}, M=lane&15
    // B layout (4x16 f32): v0 = row K(2*kh), v1 = row K(2*kh+1), N=lane&15
    // D layout (16x16 f32): comp r -> M = r + 8*kh, N = lane&15
    if (wave == 0) {
        const int m   = lane & 15;
        const int kh  = lane >> 4;
        const int wn4 = (m < 4) ? m : 0;   // clamped -> branchless weight loads
        const int wn3 = (m < 3) ? m : 0;
        v2f a, bm;
        v8f cz = {};

        // layer 1: h1 = lrelu16(v @ l1_w^T + l1_b) ; B[k][n] = l1_w[n][k]
        a.x = vstage[m * 4 + 2 * kh];
        a.y = vstage[m * 4 + 2 * kh + 1];
        {
            float wx = l1_w[wn4 * 4 + 2 * kh];
            float wy = l1_w[wn4 * 4 + 2 * kh + 1];
            bm.x = (m < 4) ? wx : 0.f;
            bm.y = (m < 4) ? wy : 0.f;
        }
        v8f d = __builtin_amdgcn_wmma_f32_16x16x4_f32(false, a, false, bm,
                                                      (short)0, cz, false, false);
        #pragma unroll
        for (int r = 0; r < 8; ++r) {
            int mm = r + 8 * kh;
            if (m < 4) hstage[mm * 4 + m] = lrelu16(d[r] + l1_b[m]);
        }

        // layer 2
        a.x = hstage[m * 4 + 2 * kh];
        a.y = hstage[m * 4 + 2 * kh + 1];
        {
            float wx = l2_w[wn4 * 4 + 2 * kh];
            float wy = l2_w[wn4 * 4 + 2 * kh + 1];
            bm.x = (m < 4) ? wx : 0.f;
            bm.y = (m < 4) ? wy : 0.f;
        }
        d = __builtin_amdgcn_wmma_f32_16x16x4_f32(false, a, false, bm,
                                                  (short)0, cz, false, false);
        #pragma unroll
        for (int r = 0; r < 8; ++r) {
            int mm = r + 8 * kh;
            if (m < 4) hstage[mm * 4 + m] = lrelu16(d[r] + l2_b[m]);
        }

        // output layer: (B,3)
        a.x = hstage[m * 4 + 2 * kh];
        a.y = hstage[m * 4 + 2 * kh + 1];
        {
            float wx = vf_w[wn3 * 4 + 2 * kh];
            float wy = vf_w[wn3 * 4 + 2 * kh + 1];
            bm.x = (m < 3) ? wx : 0.f;
            bm.y = (m < 3) ? wy : 0.f;
        }
        d = __builtin_amdgcn_wmma_f32_16x16x4_f32(false, a, false, bm,
                                                  (short)0, cz, false, false);
        float* vo = v_out + (size_t)blockIdx.x * WAVES * 3;
        #pragma unroll
        for (int r = 0; r < 8; ++r) {
            int mm = r + 8 * kh;
            if (m < 3) vo[mm * 3 + m] = d[r] + vf_b[m];
        }
    }
}

extern "C" void kernel_launch(void* const* d_in, const int* in_sizes, int n_in,
                              void* d_out, int out_size, void* d_ws, size_t ws_size,
                              hipStream_t stream) {
    (void)in_sizes; (void)n_in; (void)d_ws; (void)ws_size; (void)out_size;

    const int*   sparse = (const int*)  d_in[0];
    const int*   board  = (const int*)  d_in[1];
    const float* W_emb  = (const float*)d_in[2];
    const float* dw_w   = (const float*)d_in[3];
    const float* dw_b   = (const float*)d_in[4];
    const float* pf_w   = (const float*)d_in[5];
    const float* l1_w   = (const float*)d_in[6];
    const float* l1_b   = (const float*)d_in[7];
    const float* l2_w   = (const float*)d_in[8];
    const float* l2_b   = (const float*)d_in[9];
    const float* vf_w   = (const float*)d_in[10];
    const float* vf_b   = (const float*)d_in[11];

    float* v_out = (float*)d_out;                 // B*3 floats
    float* p_out = (float*)d_out + NBATCH * 3;    // B*225 floats

    // 212 KB dynamic LDS (CDNA5 allows up to 320 KB per workgroup)
    hipFuncSetAttribute((const void*)patnnue_fused,
                        hipFuncAttributeMaxDynamicSharedMemorySize, SMEM_BYTES);

    patnnue_fused<<<NBLK, NTHREADS, SMEM_BYTES, stream>>>(
        sparse, board, W_emb, dw_w, dw_b, pf_w,
        l1_w, l1_b, l2_w, l2_b, vf_w, vf_b, v_out, p_out);
}